// ConvolutionBlock_13477607374951
// MI455X (gfx1250) — compile-verified
//
#include <hip/hip_runtime.h>

typedef __attribute__((ext_vector_type(2))) float v2f;
typedef __attribute__((ext_vector_type(8))) float v8f;

#define HH    112
#define WW    112
#define CIN   96
#define COUT  192
#define NPIX  (HH * WW)          // 12544
#define TILE_P 64                // pixels per block (196 tiles per image)
#define PR_STRIDE 160            // floats per K-pair row (~= 64 cols * 2, padded; 160 % 64 == 32)

__global__ __launch_bounds__(256)
void conv_pw_dw_wmma(const float* __restrict__ x,
                     const float* __restrict__ wpc,
                     const float* __restrict__ wdc,
                     float* __restrict__ out)
{
    // t stored K-pair interleaved: tS[(k>>1)*PR_STRIDE + col*2 + (k&1)]
    // -> B operand (t[k][col], t[k+1][col]) is one aligned ds_load_b64.
    __shared__ float tS[(CIN / 2) * PR_STRIDE];   // 48*160*4 = 30,720 B

    const int tile = blockIdx.x;            // 0..195
    const int b    = blockIdx.y;            // 0..31
    const int p0   = tile * TILE_P;
    const int tid  = threadIdx.x;

    // Warm L2 with the pointwise weights (reused by every block).
    __builtin_prefetch(wpc + tid * 72, 0, 0);

    // 3x3 shared filter into registers (uniform address -> scalar loads).
    float f[9];
#pragma unroll
    for (int i = 0; i < 9; ++i) f[i] = wdc[i];

    // ---------------- Phase 1: depthwise 3x3 (SAME, zero pad) on input channels -> LDS
    // Branch-free: always load from a clamped (in-bounds) address, zero the weight
    // for out-of-range taps (v_cndmask instead of divergent branches).
#pragma unroll
    for (int it = 0; it < (CIN * TILE_P) / 256; ++it) {
        const int e  = tid + it * 256;
        const int ci = e >> 6;              // /64
        const int j  = e & 63;
        const int p  = p0 + j;
        const int h  = p / WW;
        const int w  = p - h * WW;
        const float* xb = x + ((size_t)(b * CIN + ci)) * NPIX;
        float acc = 0.0f;
#pragma unroll
        for (int dy = -1; dy <= 1; ++dy) {
            const int  hr   = h + dy;
            const bool rok  = (hr >= 0) && (hr < HH);
            const int  hc   = rok ? hr : h;               // clamped, always valid
#pragma unroll
            for (int dx = -1; dx <= 1; ++dx) {
                const int  wr  = w + dx;
                const bool cok = (wr >= 0) && (wr < WW);
                const int  wc  = cok ? wr : w;            // clamped, always valid
                const float v  = xb[hc * WW + wc];        // unconditional load
                const float wt = (rok && cok) ? f[(dy + 1) * 3 + (dx + 1)] : 0.0f;
                acc += wt * v;
            }
        }
        tS[(ci >> 1) * PR_STRIDE + j * 2 + (ci & 1)] = acc;
    }
    __syncthreads();

    // ---------------- Phase 2: WMMA GEMM  out[co, p] = sum_ci Wpc[co,ci] * t[ci,p]
    const int wave  = tid >> 5;
    const int lane  = tid & 31;
    const int waveM = wave >> 1;            // 0..3 -> 48 output channels each
    const int waveN = wave & 1;             // 0..1 -> 32 pixels each
    const int mBase = waveM * 48;
    const int nBase = waveN * 32;
    const int lrow  = lane & 15;
    const int khalf = lane >> 4;            // 0: lanes 0-15 (K0,K1) | 1: lanes 16-31 (K2,K3)

    v8f acc[3][2];
#pragma unroll
    for (int mt = 0; mt < 3; ++mt)
#pragma unroll
        for (int nt = 0; nt < 2; ++nt)
            acc[mt][nt] = (v8f){0.f, 0.f, 0.f, 0.f, 0.f, 0.f, 0.f, 0.f};

#pragma unroll 4
    for (int k = 0; k < CIN; k += 4) {
        // A tiles (16x4 f32): lane<16 holds (K0,K1), lane>=16 holds (K2,K3) of row mBase+lrow.
        v2f a[3];
#pragma unroll
        for (int mt = 0; mt < 3; ++mt) {
            const int row = mBase + mt * 16 + lrow;
            a[mt] = *(const v2f*)(wpc + row * CIN + k + 2 * khalf);   // 8B aligned
        }
        // B tiles (4x16 f32): single aligned ds_load_b64 per tile.
        // VGPR0 = rows {K0 | K2}, VGPR1 = rows {K1 | K3}; lanes 0-15 banks 0-31, 16-31 banks 32-63.
        const int kp = (k >> 1) + khalf;    // K-pair row: k/2 (lanes<16) or k/2+1 (lanes>=16)
        v2f bb[2];
#pragma unroll
        for (int nt = 0; nt < 2; ++nt) {
            const int col = nBase + nt * 16 + lrow;
            bb[nt] = *(const v2f*)(tS + kp * PR_STRIDE + col * 2);
        }
#pragma unroll
        for (int mt = 0; mt < 3; ++mt)
#pragma unroll
            for (int nt = 0; nt < 2; ++nt)
                acc[mt][nt] = __builtin_amdgcn_wmma_f32_16x16x4_f32(
                    /*neg_a=*/false, a[mt], /*neg_b=*/false, bb[nt],
                    /*c_mod=*/(short)0, acc[mt][nt],
                    /*reuse_a=*/false, /*reuse_b=*/false);
    }

    // ---------------- Store D tiles. C/D layout: VGPR r -> row (mtile*16 + r + 8*khalf), col = lrow.
    // Non-temporal: 308 MB write-once stream must not thrash the weight working set.
    float* ob = out + (size_t)(b * COUT) * NPIX;
#pragma unroll
    for (int mt = 0; mt < 3; ++mt) {
#pragma unroll
        for (int nt = 0; nt < 2; ++nt) {
            const int p = p0 + nBase + nt * 16 + lrow;
#pragma unroll
            for (int r = 0; r < 8; ++r) {
                const int co = mBase + mt * 16 + r + 8 * khalf;
                __builtin_nontemporal_store(acc[mt][nt][r], ob + (size_t)co * NPIX + p);
            }
        }
    }
}

extern "C" void kernel_launch(void* const* d_in, const int* in_sizes, int n_in,
                              void* d_out, int out_size, void* d_ws, size_t ws_size,
                              hipStream_t stream) {
    (void)in_sizes; (void)n_in; (void)d_ws; (void)ws_size; (void)out_size;
    const float* x   = (const float*)d_in[0];   // (32, 96, 112, 112)
    const float* wpc = (const float*)d_in[1];   // (192, 96, 1, 1)
    const float* wdc = (const float*)d_in[2];   // (1, 1, 3, 3)
    float* out = (float*)d_out;                 // (32, 192, 112, 112)

    dim3 grid(NPIX / TILE_P, 32);               // (196, 32)
    dim3 block(256);
    conv_pw_dw_wmma<<<grid, block, 0, stream>>>(x, wpc, wdc, out);
}